// Model_48026324304611
// MI455X (gfx1250) — compile-verified
//
#include <hip/hip_runtime.h>
#include <hip/hip_bf16.h>

typedef __attribute__((ext_vector_type(2))) float v2f;
typedef __attribute__((ext_vector_type(8))) float v8f;

#define N_NODES 100000
#define N_EDGES 3200000
#define F_IN    512
#define F_HID   16
#define F_OUT   64
#define M_TILES (N_NODES / 16)   // 6250, exact

// ---------------- degree / normalization ----------------

__global__ void k_init_deg(float* __restrict__ deg) {
  int i = blockIdx.x * blockDim.x + threadIdx.x;
  if (i < N_NODES) deg[i] = 1.0f;             // self-loop contribution
}

__global__ void k_deg_edges(const int* __restrict__ dst, float* __restrict__ deg) {
  int e = blockIdx.x * blockDim.x + threadIdx.x;
  if (e < N_EDGES) atomicAdd(&deg[dst[e]], 1.0f);
}

__global__ void k_rsqrt_deg(float* __restrict__ deg) {
  int i = blockIdx.x * blockDim.x + threadIdx.x;
  if (i < N_NODES) deg[i] = rsqrtf(deg[i]);   // deg >= 1 always
}

// ---------------- layer-1 GEMM: xw = x @ W1  (WMMA f32 16x16x4) ----------------
// One wave computes a 16x16 output tile (16 rows of x, all 16 hidden features).

__global__ __launch_bounds__(256) void k_gemm1_wmma(const float* __restrict__ x,
                                                    const float* __restrict__ W1,
                                                    float* __restrict__ xw) {
  __shared__ float sW[F_IN * F_HID];          // 32 KB
  for (int i = threadIdx.x; i < F_IN * F_HID; i += 256) sW[i] = W1[i];
  __syncthreads();

  const int wave = threadIdx.x >> 5;
  const int lane = threadIdx.x & 31;
  const int tile = blockIdx.x * 8 + wave;     // wave-uniform
  if (tile >= M_TILES) return;

  const int base = tile * 16;
  const int col  = lane & 15;                 // A: row index / B,C,D: column
  const int koff = (lane >> 4) * 2;           // K sub-offset per A/B layout
  const float* xr = x + (size_t)(base + col) * F_IN;

  v8f acc = {};
  #pragma unroll 8
  for (int k = 0; k < F_IN; k += 4) {
    v2f a = *(const v2f*)(xr + k + koff);     // 8B aligned
    v2f b;
    b.x = sW[(k + koff)     * F_HID + col];
    b.y = sW[(k + koff + 1) * F_HID + col];
    acc = __builtin_amdgcn_wmma_f32_16x16x4_f32(
        /*neg_a=*/false, a, /*neg_b=*/false, b,
        /*c_mod=*/(short)0, acc, /*reuse_a=*/false, /*reuse_b=*/false);
  }

  const int m0 = (lane >> 4) * 8;             // lanes 16-31 hold M = r + 8
  #pragma unroll
  for (int r = 0; r < 8; ++r)
    xw[(size_t)(base + m0 + r) * F_HID + col] = acc[r];
}

// ---------------- aggregation (16 features): acc = D^-1/2 (A+I) D^-1/2 * feat ----------------

__global__ void k_agg_self16(const float* __restrict__ feat,
                             const float* __restrict__ dis,
                             float* __restrict__ acc) {
  int i = blockIdx.x * blockDim.x + threadIdx.x;     // node*16 + f
  if (i < N_NODES * F_HID) {
    int node = i >> 4;
    float d = dis[node];
    acc[i] = d * d * feat[i];                        // self-loop term
  }
}

__global__ void k_agg_edges16(const int* __restrict__ src,
                              const int* __restrict__ dst,
                              const float* __restrict__ feat,
                              const float* __restrict__ dis,
                              float* __restrict__ acc) {
  unsigned gid = blockIdx.x * blockDim.x + threadIdx.x;  // edge*16 + f
  if (gid >= (unsigned)N_EDGES * F_HID) return;
  int e = gid >> 4;
  int f = gid & 15;
  int s = src[e];
  int d = dst[e];
  float nrm = dis[s] * dis[d];
  atomicAdd(&acc[(size_t)d * F_HID + f], nrm * feat[(size_t)s * F_HID + f]);
}

__global__ void k_bias_relu16(const float* __restrict__ acc,
                              const float* __restrict__ bias,
                              float* __restrict__ h) {
  int i = blockIdx.x * blockDim.x + threadIdx.x;
  if (i < N_NODES * F_HID) h[i] = fmaxf(acc[i] + bias[i & 15], 0.0f);
}

// ---------------- layer-2 GEMM: out = relu(a2 @ W2 + b2)  (WMMA f32) ----------------
// One wave: 16 rows x 64 cols = four 16x16 C tiles, K = 16 (4 WMMA steps each).

__global__ __launch_bounds__(256) void k_gemm2_wmma(const float* __restrict__ a2,
                                                    const float* __restrict__ W2,
                                                    const float* __restrict__ b2,
                                                    float* __restrict__ out) {
  __shared__ float sW[F_HID * F_OUT];         // 4 KB
  __shared__ float sB[F_OUT];
  for (int i = threadIdx.x; i < F_HID * F_OUT; i += 256) sW[i] = W2[i];
  if (threadIdx.x < F_OUT) sB[threadIdx.x] = b2[threadIdx.x];
  __syncthreads();

  const int wave = threadIdx.x >> 5;
  const int lane = threadIdx.x & 31;
  const int tile = blockIdx.x * 8 + wave;
  if (tile >= M_TILES) return;

  const int base = tile * 16;
  const int col  = lane & 15;
  const int koff = (lane >> 4) * 2;
  const float* ar = a2 + (size_t)(base + col) * F_HID;

  v8f acc[4] = {{}, {}, {}, {}};
  #pragma unroll
  for (int k = 0; k < F_HID; k += 4) {
    v2f a = *(const v2f*)(ar + k + koff);
    #pragma unroll
    for (int j = 0; j < 4; ++j) {
      v2f b;
      b.x = sW[(k + koff)     * F_OUT + j * 16 + col];
      b.y = sW[(k + koff + 1) * F_OUT + j * 16 + col];
      acc[j] = __builtin_amdgcn_wmma_f32_16x16x4_f32(
          false, a, false, b, (short)0, acc[j], false, false);
    }
  }

  const int m0 = (lane >> 4) * 8;
  #pragma unroll
  for (int j = 0; j < 4; ++j)
    #pragma unroll
    for (int r = 0; r < 8; ++r)
      out[(size_t)(base + m0 + r) * F_OUT + j * 16 + col] =
          fmaxf(acc[j][r] + sB[j * 16 + col], 0.0f);
}

// ---------------- launcher ----------------

extern "C" void kernel_launch(void* const* d_in, const int* in_sizes, int n_in,
                              void* d_out, int out_size, void* d_ws, size_t ws_size,
                              hipStream_t stream) {
  const float* x  = (const float*)d_in[0];
  const int*   ei = (const int*)d_in[1];      // [2, N_EDGES] int32
  const float* W1 = (const float*)d_in[2];
  const float* b1 = (const float*)d_in[3];
  const float* W2 = (const float*)d_in[4];
  const float* b2 = (const float*)d_in[5];
  float* out = (float*)d_out;

  const int* src = ei;
  const int* dst = ei + N_EDGES;

  // workspace layout: dis (n) | bufA (n*16) | bufB (n*16)
  char* ws = (char*)d_ws;
  float* dis  = (float*)ws;                                   // 400 KB
  float* bufA = (float*)(ws + ((size_t)N_NODES * 4 + 511 & ~511ull) + 0);
  bufA = (float*)(ws + 512 * (((size_t)N_NODES * 4 + 511) / 512));
  float* bufB = bufA + (size_t)N_NODES * F_HID;

  const int B = 256;
  const int gN   = (N_NODES + B - 1) / B;                     // 391
  const int gE   = (N_EDGES + B - 1) / B;                     // 12500
  const int gNF  = (N_NODES * F_HID + B - 1) / B;             // 6250
  const int gEF  = (int)(((size_t)N_EDGES * F_HID + B - 1) / B); // 200000
  const int gMM  = (M_TILES + 7) / 8;                         // 782 blocks, 8 waves each

  // normalization: dis = rsqrt(1 + in_degree)
  k_init_deg<<<gN, B, 0, stream>>>(dis);
  k_deg_edges<<<gE, B, 0, stream>>>(dst, dis);
  k_rsqrt_deg<<<gN, B, 0, stream>>>(dis);

  // layer 1: xw = x @ W1  -> bufA
  k_gemm1_wmma<<<gMM, B, 0, stream>>>(x, W1, bufA);

  // agg1 -> bufB, then h = relu(bufB + b1) -> bufA (reuse)
  k_agg_self16<<<gNF, B, 0, stream>>>(bufA, dis, bufB);
  k_agg_edges16<<<gEF, B, 0, stream>>>(src, dst, bufA, dis, bufB);
  k_bias_relu16<<<gNF, B, 0, stream>>>(bufB, b1, bufA);

  // agg2 on h (16-dim, before GEMM2 since Â(HW)=(ÂH)W) -> bufB
  k_agg_self16<<<gNF, B, 0, stream>>>(bufA, dis, bufB);
  k_agg_edges16<<<gEF, B, 0, stream>>>(src, dst, bufA, dis, bufB);

  // layer 2: out = relu(bufB @ W2 + b2)
  k_gemm2_wmma<<<gMM, B, 0, stream>>>(bufB, W2, b2, out);

  (void)in_sizes; (void)n_in; (void)out_size; (void)ws_size;
}